// SoftmaxCrossEntropyOHEMLoss_4612794875978
// MI455X (gfx1250) — compile-verified
//
#include <hip/hip_runtime.h>
#include <float.h>

// ---------------------------------------------------------------------------
// OHEM softmax cross-entropy, MI455X (gfx1250, wave32).
// Memory-bound: ~190MB HBM traffic -> ~8us at 23.3 TB/s. No GEMM -> no WMMA.
// Compute pass stages the 19-channel logit slab into LDS with the CDNA5
// Tensor Data Mover (tensor_load_to_lds + s_wait_tensorcnt), which removes
// the 76-VGPR register staging and raises occupancy for the HBM stream.
// ---------------------------------------------------------------------------

#define IGNORE_LABEL (-1)
#define THRESH_F 0.7f
#define MIN_KEPT 256
#define N_ 4
#define C_ 19
#define H_ 512
#define W_ 1024
#define HW_ (H_ * W_)
#define P_ (N_ * HW_)   /* 2,097,152 pixels */
#define PV4_ (P_ / 4)   /* 524,288 vec4 groups */

#define TILE_W 512      /* pixels per block tile */
#define TPB_C 128       /* threads per compute block (4 px / thread) */

// control block layout (uint32 words) inside workspace
#define CT_NVALID 0
#define CT_SEL1   1
#define CT_RANK1  2
#define CT_SEL2   3
#define CT_RANK2  4
#define CT_THR    6
#define CT_SUMN   8
#define CT_SUMD   9
#define CT_BINS1  16
#define CT_BINS2  (16 + 2048)
#define CT_BINS3  (16 + 4096)
#define CT_WORDS  (16 + 4096 + 1024)

typedef unsigned tdm_v4u __attribute__((ext_vector_type(4)));
typedef int      tdm_v4i __attribute__((ext_vector_type(4)));
typedef int      tdm_v8i __attribute__((ext_vector_type(8)));

__global__ void ohem_zero(unsigned* __restrict__ ctrl) {
  int i = blockIdx.x * blockDim.x + threadIdx.x;
  if (i < CT_WORDS) ctrl[i] = 0u;
}

// Pass 1: TDM-stage a [C_ x TILE_W] slab of logits into LDS, then per-pixel
// log-softmax; store gt-prob bits (+inf for invalid) and cw[label]*nll.
__global__ __launch_bounds__(TPB_C) void ohem_compute(
    const float* __restrict__ predict, const int* __restrict__ target,
    const float* __restrict__ cw, unsigned* __restrict__ predBits,
    float* __restrict__ wnll, unsigned* __restrict__ ctrl) {
  __shared__ float tile[C_ * TILE_W];   // 19*512*4 = 38,912 B
  __shared__ unsigned s_nv;
  if (threadIdx.x == 0) s_nv = 0u;

  const long p0  = (long)blockIdx.x * TILE_W;   // first pixel of tile
  const int  n   = (int)(p0 / HW_);
  const int  hw0 = (int)(p0 - (long)n * HW_);

  if (threadIdx.x == 0) {
    // ---- build Tensor DMA descriptor (D#) ----
    const float* gp = predict + (long)n * C_ * HW_ + hw0;  // tile start
    unsigned long long gaddr = (unsigned long long)(uintptr_t)gp;
    unsigned ldsoff = (unsigned)(unsigned long long)(uintptr_t)&tile[0];

    tdm_v4u g0;
    g0[0] = 1u;                                   // count=1, no gather
    g0[1] = ldsoff;                               // lds_addr (bytes)
    g0[2] = (unsigned)(gaddr & 0xFFFFFFFFull);    // global_addr[31:0]
    g0[3] = (unsigned)((gaddr >> 32) & 0x1FFFFFFull) | (2u << 30); // [56:32] | type=2

    tdm_v8i g1;
    g1[0] = (int)(2u << 16);                      // wg_mask=0, data_size=4B
    g1[1] = (int)(((unsigned)TILE_W & 0xFFFFu) << 16);   // tensor_dim0 lo16
    g1[2] = (int)(((unsigned)TILE_W >> 16) | ((unsigned)C_ << 16)); // d0 hi | tensor_dim1 lo
    g1[3] = (int)(((unsigned)C_ >> 16) | ((unsigned)TILE_W << 16)); // d1 hi | tile_dim0
    g1[4] = (int)((unsigned)C_);                  // tile_dim1=19, tile_dim2=0 -> 2D
    g1[5] = (int)((unsigned)HW_);                 // tensor_dim0_stride lo32
    g1[6] = 0;                                    // stride0 hi16 | stride1 lo16
    g1[7] = 0;                                    // stride1 hi32

    tdm_v4i g2 = {0, 0, 0, 0};
    tdm_v4i g3 = {0, 0, 0, 0};

#if defined(__clang_major__) && __clang_major__ >= 23
    tdm_v8i g4 = {0, 0, 0, 0, 0, 0, 0, 0};
    __builtin_amdgcn_tensor_load_to_lds(g0, g1, g2, g3, g4, 0);
#else
    __builtin_amdgcn_tensor_load_to_lds(g0, g1, g2, g3, 0);
#endif
    __builtin_amdgcn_s_wait_tensorcnt(0);
  }
  __syncthreads();   // slab is in LDS; all waves may consume

  const int  tid = threadIdx.x;
  const int  x0  = tid * 4;                 // pixel offset within tile
  const long p4  = p0 + x0;                 // global pixel index

  const int4 lab4 = *reinterpret_cast<const int4*>(target + p4);
  int lab[4] = {lab4.x, lab4.y, lab4.z, lab4.w};

  bool valv[4]; int safe[4];
#pragma unroll
  for (int j = 0; j < 4; ++j) {
    valv[j] = (lab[j] != IGNORE_LABEL);
    safe[j] = valv[j] ? lab[j] : 0;
  }

  float mx[4], gt[4];
#pragma unroll
  for (int j = 0; j < 4; ++j) { mx[j] = -FLT_MAX; gt[j] = 0.f; }

  // sweep A: max + gt-logit capture (ds_load_b128 per channel row)
#pragma unroll
  for (int c = 0; c < C_; ++c) {
    float4 v = *reinterpret_cast<const float4*>(&tile[c * TILE_W + x0]);
    float e[4] = {v.x, v.y, v.z, v.w};
#pragma unroll
    for (int j = 0; j < 4; ++j) {
      mx[j] = fmaxf(mx[j], e[j]);
      if (safe[j] == c) gt[j] = e[j];
    }
  }

  // sweep B: sum of exp (re-read LDS; trades LDS BW for VGPRs)
  float sm[4] = {0.f, 0.f, 0.f, 0.f};
#pragma unroll
  for (int c = 0; c < C_; ++c) {
    float4 v = *reinterpret_cast<const float4*>(&tile[c * TILE_W + x0]);
    float e[4] = {v.x, v.y, v.z, v.w};
#pragma unroll
    for (int j = 0; j < 4; ++j) sm[j] += __expf(e[j] - mx[j]);
  }

  unsigned pb[4]; float wl[4];
#pragma unroll
  for (int j = 0; j < 4; ++j) {
    float logsum = __logf(sm[j]) + mx[j];
    float nll    = logsum - gt[j];
    float prob   = __expf(-nll);                            // exp(logp_gt)
    pb[j] = valv[j] ? __float_as_uint(prob) : 0x7F800000u;  // +inf sorts last
    wl[j] = valv[j] ? cw[safe[j]] * nll : 0.f;
  }
  *reinterpret_cast<uint4*>(predBits + p4) = make_uint4(pb[0], pb[1], pb[2], pb[3]);
  *reinterpret_cast<float4*>(wnll + p4)    = make_float4(wl[0], wl[1], wl[2], wl[3]);

  unsigned cnt = (unsigned)valv[0] + (unsigned)valv[1] +
                 (unsigned)valv[2] + (unsigned)valv[3];
  atomicAdd(&s_nv, cnt);
  __syncthreads();
  if (threadIdx.x == 0) atomicAdd(&ctrl[CT_NVALID], s_nv);
}

// Radix-select histograms over float bit patterns (non-negative floats:
// uint order == float order). level 0: bits[31:21]; 1: bits[20:10]; 2: bits[9:0].
__global__ __launch_bounds__(256) void ohem_hist(
    const unsigned* __restrict__ predBits, unsigned* __restrict__ ctrl, int level) {
  __shared__ unsigned lb[2048];
  const int nb = (level == 2) ? 1024 : 2048;
  unsigned* gb = ctrl + ((level == 0) ? CT_BINS1 : (level == 1) ? CT_BINS2 : CT_BINS3);
  for (int i = threadIdx.x; i < nb; i += blockDim.x) lb[i] = 0u;
  __syncthreads();
  const unsigned sel1 = ctrl[CT_SEL1];
  const unsigned sel2 = ctrl[CT_SEL2];
  const uint4* pv = reinterpret_cast<const uint4*>(predBits);
  const int stride = gridDim.x * blockDim.x;
  for (int i = blockIdx.x * blockDim.x + threadIdx.x; i < PV4_; i += stride) {
    __builtin_prefetch(pv + i + stride, 0, 0);   // global_prefetch_b8
    uint4 q = pv[i];
    unsigned b[4] = {q.x, q.y, q.z, q.w};
#pragma unroll
    for (int j = 0; j < 4; ++j) {
      unsigned bits = b[j];
      if (level == 0) {
        atomicAdd(&lb[bits >> 21], 1u);
      } else if (level == 1) {
        if ((bits >> 21) == sel1) atomicAdd(&lb[(bits >> 10) & 0x7FFu], 1u);
      } else {
        if (((bits >> 21) == sel1) && (((bits >> 10) & 0x7FFu) == sel2))
          atomicAdd(&lb[bits & 0x3FFu], 1u);
      }
    }
  }
  __syncthreads();
  for (int i = threadIdx.x; i < nb; i += blockDim.x)
    if (lb[i]) atomicAdd(&gb[i], lb[i]);
}

// Single-thread scans: find the bin holding rank k at each radix level, then
// reconstruct kth value and the OHEM threshold.
__global__ void ohem_select(unsigned* __restrict__ ctrl, int level) {
  if (blockIdx.x != 0 || threadIdx.x != 0) return;
  if (level == 0) {
    long nv = (long)ctrl[CT_NVALID];
    long k  = ((long)MIN_KEPT < nv ? (long)MIN_KEPT : nv) - 1;
    if (k < 0) k = 0;
    if (k > (long)P_ - 1) k = (long)P_ - 1;
    unsigned long long cum = 0; unsigned sel = 2047, rank = 0;
    for (int i = 0; i < 2048; ++i) {
      unsigned c = ctrl[CT_BINS1 + i];
      if (cum + c > (unsigned long long)k) { sel = (unsigned)i; rank = (unsigned)((unsigned long long)k - cum); break; }
      cum += c;
    }
    ctrl[CT_SEL1] = sel; ctrl[CT_RANK1] = rank;
  } else if (level == 1) {
    unsigned k = ctrl[CT_RANK1];
    unsigned cum = 0; unsigned sel = 2047, rank = 0;
    for (int i = 0; i < 2048; ++i) {
      unsigned c = ctrl[CT_BINS2 + i];
      if (cum + c > k) { sel = (unsigned)i; rank = k - cum; break; }
      cum += c;
    }
    ctrl[CT_SEL2] = sel; ctrl[CT_RANK2] = rank;
  } else {
    unsigned k = ctrl[CT_RANK2];
    unsigned cum = 0; unsigned low = 1023;
    for (int i = 0; i < 1024; ++i) {
      unsigned c = ctrl[CT_BINS3 + i];
      if (cum + c > k) { low = (unsigned)i; break; }
      cum += c;
    }
    unsigned kbits = (ctrl[CT_SEL1] << 21) | (ctrl[CT_SEL2] << 10) | low;
    float kth = __uint_as_float(kbits);
    long nv = (long)ctrl[CT_NVALID];
    float thr;
    if ((long)MIN_KEPT >= nv) {
      thr = FLT_MAX;                 // keep all valid (invalid carry +inf)
    } else {
      thr = (kth > THRESH_F) ? kth : THRESH_F;   // MIN_KEPT > 0 is constant
    }
    ctrl[CT_THR] = __float_as_uint(thr);
  }
}

// Final weighted reduction over kept pixels.
__global__ __launch_bounds__(256) void ohem_final(
    const unsigned* __restrict__ predBits, const float* __restrict__ wnll,
    const int* __restrict__ target, const float* __restrict__ cw,
    unsigned* __restrict__ ctrl) {
  __shared__ float rN[256], rD[256];
  __shared__ float scw[C_];
  for (int i = threadIdx.x; i < C_; i += blockDim.x) scw[i] = cw[i];
  __syncthreads();
  const float thr = __uint_as_float(ctrl[CT_THR]);
  float num = 0.f, den = 0.f;
  const uint4*  pv = reinterpret_cast<const uint4*>(predBits);
  const float4* wv = reinterpret_cast<const float4*>(wnll);
  const int4*   tv = reinterpret_cast<const int4*>(target);
  const int stride = gridDim.x * blockDim.x;
  for (int i = blockIdx.x * blockDim.x + threadIdx.x; i < PV4_; i += stride) {
    __builtin_prefetch(pv + i + stride, 0, 0);
    uint4 q = pv[i]; float4 w = wv[i]; int4 tg = tv[i];
    unsigned qb[4] = {q.x, q.y, q.z, q.w};
    float    wb[4] = {w.x, w.y, w.z, w.w};
    int      tb[4] = {tg.x, tg.y, tg.z, tg.w};
#pragma unroll
    for (int j = 0; j < 4; ++j) {
      float p = __uint_as_float(qb[j]);          // +inf for invalid -> excluded
      if (p <= thr) {
        int l = tb[j];
        l = (l < 0 || l >= C_) ? 0 : l;
        num += wb[j];
        den += scw[l];
      }
    }
  }
  rN[threadIdx.x] = num; rD[threadIdx.x] = den;
  __syncthreads();
  for (int s = 128; s > 0; s >>= 1) {
    if ((int)threadIdx.x < s) {
      rN[threadIdx.x] += rN[threadIdx.x + s];
      rD[threadIdx.x] += rD[threadIdx.x + s];
    }
    __syncthreads();
  }
  if (threadIdx.x == 0) {
    atomicAdd(reinterpret_cast<float*>(ctrl) + CT_SUMN, rN[0]);  // global_atomic_add_f32
    atomicAdd(reinterpret_cast<float*>(ctrl) + CT_SUMD, rD[0]);
  }
}

__global__ void ohem_div(const unsigned* __restrict__ ctrl, float* __restrict__ out) {
  if (blockIdx.x == 0 && threadIdx.x == 0) {
    const float* f = reinterpret_cast<const float*>(ctrl);
    out[0] = f[CT_SUMN] / f[CT_SUMD];
  }
}

extern "C" void kernel_launch(void* const* d_in, const int* in_sizes, int n_in,
                              void* d_out, int out_size, void* d_ws, size_t ws_size,
                              hipStream_t stream) {
  (void)in_sizes; (void)n_in; (void)out_size; (void)ws_size;
  const float* predict = (const float*)d_in[0];
  const int*   target  = (const int*)d_in[1];
  const float* cw      = (const float*)d_in[2];
  float* out = (float*)d_out;

  // workspace: [ predBits: P_ u32 | wnll: P_ f32 | ctrl: CT_WORDS u32 ]
  unsigned* predBits = (unsigned*)d_ws;
  float*    wnll     = (float*)((char*)d_ws + (size_t)P_ * 4);
  unsigned* ctrl     = (unsigned*)((char*)d_ws + (size_t)P_ * 8);

  ohem_zero<<<(CT_WORDS + 255) / 256, 256, 0, stream>>>(ctrl);
  ohem_compute<<<P_ / TILE_W, TPB_C, 0, stream>>>(predict, target, cw, predBits, wnll, ctrl);
  ohem_hist<<<1024, 256, 0, stream>>>(predBits, ctrl, 0);
  ohem_select<<<1, 1, 0, stream>>>(ctrl, 0);
  ohem_hist<<<1024, 256, 0, stream>>>(predBits, ctrl, 1);
  ohem_select<<<1, 1, 0, stream>>>(ctrl, 1);
  ohem_hist<<<1024, 256, 0, stream>>>(predBits, ctrl, 2);
  ohem_select<<<1, 1, 0, stream>>>(ctrl, 2);
  ohem_final<<<1024, 256, 0, stream>>>(predBits, wnll, target, cw, ctrl);
  ohem_div<<<1, 1, 0, stream>>>(ctrl, out);
}